// GCN_34626026340524
// MI455X (gfx1250) — compile-verified
//
#include <hip/hip_runtime.h>
#include <hip/hip_bf16.h>

typedef __attribute__((ext_vector_type(2))) float v2f;
typedef __attribute__((ext_vector_type(8))) float v8f;

#define DH 512
#define LDA 516   // padded LDS row stride: 516 % 64 == 4 -> conflict-free column reads

// ---------------------------------------------------------------------------
// small utility kernels
// ---------------------------------------------------------------------------
__global__ void __launch_bounds__(256) zero_f_k(float* p, int n) {
    int i = blockIdx.x * 256 + threadIdx.x;
    if (i < n) p[i] = 0.0f;
}
__global__ void __launch_bounds__(256) zero_i_k(int* p, int n) {
    int i = blockIdx.x * 256 + threadIdx.x;
    if (i < n) p[i] = 0;
}

__global__ void __launch_bounds__(256) deg_k(const int* __restrict__ col,
                                             const float* __restrict__ ew,
                                             float* __restrict__ deg, int e) {
    int i = blockIdx.x * 256 + threadIdx.x;
    if (i < e) atomicAdd(&deg[col[i]], ew[i]);
}

__global__ void __launch_bounds__(256) dinv_k(float* d, int n) {
    int i = blockIdx.x * 256 + threadIdx.x;
    if (i < n) d[i] = rsqrtf(d[i] + 1.0f);   // deg+1 > 0 always
}

// ---------------------------------------------------------------------------
// CSR construction: counts -> exclusive scan -> fill (row, coef) per dst node
// ---------------------------------------------------------------------------
__global__ void __launch_bounds__(256) count_k(const int* __restrict__ col,
                                               int* __restrict__ counts, int e) {
    int i = blockIdx.x * 256 + threadIdx.x;
    if (i < e) atomicAdd(&counts[col[i]], 1);
}

__global__ void __launch_bounds__(256)
scan_block_k(const int* __restrict__ counts, int* __restrict__ offs,
             int* __restrict__ bsums, int n) {
    __shared__ int sm[256];
    int t = threadIdx.x;
    int i = blockIdx.x * 256 + t;
    int v = (i < n) ? counts[i] : 0;
    sm[t] = v;
    __syncthreads();
    #pragma unroll
    for (int off = 1; off < 256; off <<= 1) {      // Hillis-Steele inclusive
        int x = (t >= off) ? sm[t - off] : 0;
        __syncthreads();
        sm[t] += x;
        __syncthreads();
    }
    if (i < n) offs[i] = sm[t] - v;                // exclusive within block
    if (t == 255) bsums[blockIdx.x] = sm[255];
}

__global__ void __launch_bounds__(256) scan_top_k(int* bsums, int nb) {
    __shared__ int sm[256];
    int t = threadIdx.x;
    int v = (t < nb) ? bsums[t] : 0;
    sm[t] = v;
    __syncthreads();
    #pragma unroll
    for (int off = 1; off < 256; off <<= 1) {
        int x = (t >= off) ? sm[t - off] : 0;
        __syncthreads();
        sm[t] += x;
        __syncthreads();
    }
    if (t < nb) bsums[t] = sm[t] - v;              // exclusive block bases
}

__global__ void __launch_bounds__(256)
scan_add_k(int* __restrict__ offs, const int* __restrict__ bsums, int n) {
    int i = blockIdx.x * 256 + threadIdx.x;
    if (i < n) offs[i] += bsums[blockIdx.x];
}

__global__ void __launch_bounds__(256)
fill_k(const int* __restrict__ row, const int* __restrict__ col,
       const float* __restrict__ ew, const float* __restrict__ dinv,
       const int* __restrict__ offs, int* __restrict__ cursor,
       int* __restrict__ csr_row, float* __restrict__ csr_coef, int e) {
    int i = blockIdx.x * 256 + threadIdx.x;
    if (i < e) {
        int r = row[i], c = col[i];
        int p = offs[c] + atomicAdd(&cursor[c], 1);
        csr_row[p]  = r;
        csr_coef[p] = dinv[r] * ew[i] * dinv[c];
    }
}

// ---------------------------------------------------------------------------
// GEMM: H[N x 512] = X[N x 512] * W[512 x 512]   (fp32 WMMA 16x16x4)
// One block = 16 output rows x all 512 cols. 8 waves, 4 N-tiles per wave.
// A tile staged into LDS via gfx1250 async global->LDS copies.
// ---------------------------------------------------------------------------
__global__ void __launch_bounds__(256)
gemm_wmma_f32(const float* __restrict__ X, const float* __restrict__ W,
              float* __restrict__ H) {
    __shared__ __align__(16) float As[16 * LDA];

    const int tid = threadIdx.x;
    const int m0  = blockIdx.x * 16;

    // async copy of X[m0:m0+16, :] into LDS: 2048 x 16B, one b128 per lane-iter
    {
        const char* Xg = reinterpret_cast<const char*>(X + (size_t)m0 * DH);
        unsigned lds_base = (unsigned)(uintptr_t)(&As[0]);   // low 32b = LDS offset
        #pragma unroll
        for (int it = 0; it < 8; ++it) {
            int i = tid + it * 256;                 // float4 index in [0, 2048)
            int r = (i * 4) / DH;
            int c = (i * 4) % DH;
            unsigned lds_off = lds_base + (unsigned)(r * LDA + c) * 4u;
            unsigned long long ga = (unsigned long long)(Xg + (size_t)i * 16);
            asm volatile("global_load_async_to_lds_b128 %0, %1, off"
                         :: "v"(lds_off), "v"(ga) : "memory");
        }
        asm volatile("s_wait_asynccnt 0x0" ::: "memory");
    }
    __syncthreads();

    const int lane = tid & 31;
    const int wave = tid >> 5;
    const int lo   = lane & 15;
    const int hi   = lane >> 4;
    const int nbase = wave * 64;               // this wave's 4 N-tiles

    v8f acc0 = {}, acc1 = {}, acc2 = {}, acc3 = {};

    for (int k0 = 0; k0 < DH; k0 += 4) {
        // A fragment: A[M=lo][K = v + 2*hi]
        const float* ap = &As[lo * LDA + k0 + 2 * hi];
        v2f a; a.x = ap[0]; a.y = ap[1];

        // B fragments: B[K = v + 2*hi][N = lo] = W[(k0 + v + 2*hi)][n0 + lo]
        const float* wp = W + (size_t)(k0 + 2 * hi) * DH + lo;
        __builtin_prefetch(wp + 4 * DH, 0, 0);     // next K-step rows
        v2f b0, b1, b2, b3;
        b0.x = wp[nbase +  0]; b0.y = wp[DH + nbase +  0];
        b1.x = wp[nbase + 16]; b1.y = wp[DH + nbase + 16];
        b2.x = wp[nbase + 32]; b2.y = wp[DH + nbase + 32];
        b3.x = wp[nbase + 48]; b3.y = wp[DH + nbase + 48];

        acc0 = __builtin_amdgcn_wmma_f32_16x16x4_f32(false, a, false, b0, (short)0, acc0, false, false);
        acc1 = __builtin_amdgcn_wmma_f32_16x16x4_f32(false, a, false, b1, (short)0, acc1, false, false);
        acc2 = __builtin_amdgcn_wmma_f32_16x16x4_f32(false, a, false, b2, (short)0, acc2, false, false);
        acc3 = __builtin_amdgcn_wmma_f32_16x16x4_f32(false, a, false, b3, (short)0, acc3, false, false);
    }

    // D layout: D[M = v + 8*hi][N = lo]
    #pragma unroll
    for (int v = 0; v < 8; ++v) {
        int m = m0 + v + 8 * hi;
        float* hp = H + (size_t)m * DH + nbase + lo;
        hp[ 0] = acc0[v];
        hp[16] = acc1[v];
        hp[32] = acc2[v];
        hp[48] = acc3[v];
    }
}

// ---------------------------------------------------------------------------
// Gather-aggregate (wave per destination node), fused bias (+ optional ReLU):
//   out[i,:] = act( dinv[i]^2 * h[i,:] + sum_e coef_e * h[src_e,:] + bias )
// ---------------------------------------------------------------------------
__global__ void __launch_bounds__(256)
gather_k(const float* __restrict__ H, const int* __restrict__ csr_row,
         const float* __restrict__ csr_coef, const int* __restrict__ offs,
         const int* __restrict__ counts, const float* __restrict__ dinv,
         const float* __restrict__ bias, float* __restrict__ OUT,
         int n, int do_relu) {
    int wid  = (int)((blockIdx.x * 256 + threadIdx.x) >> 5);
    int lane = threadIdx.x & 31;
    if (wid >= n) return;

    float s  = dinv[wid];
    float s2 = s * s;

    float4 acc[4];
    const float* hp = H + (size_t)wid * DH;
    #pragma unroll
    for (int j = 0; j < 4; ++j) {
        float4 v = *reinterpret_cast<const float4*>(hp + lane * 4 + j * 128);
        acc[j].x = v.x * s2; acc[j].y = v.y * s2;
        acc[j].z = v.z * s2; acc[j].w = v.w * s2;
    }

    int beg = offs[wid];
    int cnt = counts[wid];
    for (int t = 0; t < cnt; ++t) {
        int   r  = csr_row[beg + t];
        float cf = csr_coef[beg + t];
        const float* hr = H + (size_t)r * DH;
        #pragma unroll
        for (int j = 0; j < 4; ++j) {
            float4 v = *reinterpret_cast<const float4*>(hr + lane * 4 + j * 128);
            acc[j].x = fmaf(cf, v.x, acc[j].x);
            acc[j].y = fmaf(cf, v.y, acc[j].y);
            acc[j].z = fmaf(cf, v.z, acc[j].z);
            acc[j].w = fmaf(cf, v.w, acc[j].w);
        }
    }

    float* op = OUT + (size_t)wid * DH;
    #pragma unroll
    for (int j = 0; j < 4; ++j) {
        int d = lane * 4 + j * 128;
        float4 o;
        o.x = acc[j].x + bias[d + 0];
        o.y = acc[j].y + bias[d + 1];
        o.z = acc[j].z + bias[d + 2];
        o.w = acc[j].w + bias[d + 3];
        if (do_relu) {
            o.x = fmaxf(o.x, 0.0f); o.y = fmaxf(o.y, 0.0f);
            o.z = fmaxf(o.z, 0.0f); o.w = fmaxf(o.w, 0.0f);
        }
        *reinterpret_cast<float4*>(op + d) = o;
    }
}

// ---------------------------------------------------------------------------
// layer 3: per-node dot with W3[512x1] (wave per node), then scalar gather
// ---------------------------------------------------------------------------
__global__ void __launch_bounds__(256)
rowdot_k(const float* __restrict__ H, const float* __restrict__ W3,
         float* __restrict__ h3, int n) {
    int wid  = (int)((blockIdx.x * 256 + threadIdx.x) >> 5);
    int lane = threadIdx.x & 31;
    if (wid >= n) return;
    const float* hp = H + (size_t)wid * DH;
    float s = 0.0f;
    #pragma unroll
    for (int t = 0; t < DH / 32; ++t) {
        int j = lane + t * 32;
        s = fmaf(hp[j], W3[j], s);
    }
    #pragma unroll
    for (int off = 16; off > 0; off >>= 1) s += __shfl_down(s, off, 32);
    if (lane == 0) h3[wid] = s;
}

__global__ void __launch_bounds__(256)
gather_scalar_k(const float* __restrict__ h3, const int* __restrict__ csr_row,
                const float* __restrict__ csr_coef, const int* __restrict__ offs,
                const int* __restrict__ counts, const float* __restrict__ dinv,
                const float* __restrict__ b3, float* __restrict__ out, int n) {
    int i = blockIdx.x * 256 + threadIdx.x;
    if (i < n) {
        float s = dinv[i];
        float acc = h3[i] * s * s;
        int beg = offs[i], cnt = counts[i];
        for (int t = 0; t < cnt; ++t)
            acc = fmaf(csr_coef[beg + t], h3[csr_row[beg + t]], acc);
        out[i] = acc + b3[0];
    }
}

// ---------------------------------------------------------------------------
extern "C" void kernel_launch(void* const* d_in, const int* in_sizes, int n_in,
                              void* d_out, int out_size, void* d_ws, size_t ws_size,
                              hipStream_t stream) {
    const float* x  = (const float*)d_in[0];
    const int*   ei = (const int*)  d_in[1];
    const float* ew = (const float*)d_in[2];
    const float* W1 = (const float*)d_in[3];
    const float* b1 = (const float*)d_in[4];
    const float* W2 = (const float*)d_in[5];
    const float* b2 = (const float*)d_in[6];
    const float* W3 = (const float*)d_in[7];
    const float* b3 = (const float*)d_in[8];
    float* out = (float*)d_out;

    const int N_ = in_sizes[0] / DH;          // 50000
    const int E_ = in_sizes[2];               // 800000
    const int* row = ei;
    const int* col = ei + E_;

    // workspace carve (256B-aligned chunks)
    char* w = (char*)d_ws;
    auto carve = [&](size_t bytes) {
        char* p = w;
        w += (bytes + 255) & ~(size_t)255;
        return p;
    };
    float* dinv     = (float*)carve((size_t)N_ * 4);
    float* h3       = (float*)carve((size_t)N_ * 4);
    int*   counts   = (int*)  carve((size_t)N_ * 4);
    int*   offs     = (int*)  carve((size_t)N_ * 4);
    int*   cursor   = (int*)  carve((size_t)N_ * 4);
    int*   bsums    = (int*)  carve(256 * 4);
    int*   csr_row  = (int*)  carve((size_t)E_ * 4);
    float* csr_coef = (float*)carve((size_t)E_ * 4);
    float* bufA     = (float*)carve((size_t)N_ * DH * 4);
    float* bufB     = (float*)carve((size_t)N_ * DH * 4);

    const int gN  = (N_ + 255) / 256;         // 196 blocks (<= 256 for top scan)
    const int gE  = (E_ + 255) / 256;
    const int gNW = (N_ + 7) / 8;             // wave per node
    const int gMM = N_ / 16;                  // 16-row stripes (N % 16 == 0)

    // normalization: dinv = rsqrt(deg + 1)
    zero_f_k<<<gN, 256, 0, stream>>>(dinv, N_);
    deg_k   <<<gE, 256, 0, stream>>>(col, ew, dinv, E_);
    dinv_k  <<<gN, 256, 0, stream>>>(dinv, N_);

    // CSR build (once, reused by all 3 layers)
    zero_i_k    <<<gN, 256, 0, stream>>>(counts, N_);
    count_k     <<<gE, 256, 0, stream>>>(col, counts, E_);
    scan_block_k<<<gN, 256, 0, stream>>>(counts, offs, bsums, N_);
    scan_top_k  <<<1,  256, 0, stream>>>(bsums, gN);
    scan_add_k  <<<gN, 256, 0, stream>>>(offs, bsums, N_);
    zero_i_k    <<<gN, 256, 0, stream>>>(cursor, N_);
    fill_k      <<<gE, 256, 0, stream>>>(row, col, ew, dinv, offs, cursor,
                                         csr_row, csr_coef, E_);

    // layer 1
    gemm_wmma_f32<<<gMM, 256, 0, stream>>>(x, W1, bufA);
    gather_k     <<<gNW, 256, 0, stream>>>(bufA, csr_row, csr_coef, offs, counts,
                                           dinv, b1, bufB, N_, /*relu=*/1);
    // layer 2
    gemm_wmma_f32<<<gMM, 256, 0, stream>>>(bufB, W2, bufA);
    gather_k     <<<gNW, 256, 0, stream>>>(bufA, csr_row, csr_coef, offs, counts,
                                           dinv, b2, bufB, N_, /*relu=*/1);
    // layer 3
    rowdot_k       <<<gNW, 256, 0, stream>>>(bufB, W3, h3, N_);
    gather_scalar_k<<<gN,  256, 0, stream>>>(h3, csr_row, csr_coef, offs, counts,
                                             dinv, b3, out, N_);
}